// AdaptiveEMA_34144990003442
// MI455X (gfx1250) — compile-verified
//
#include <hip/hip_runtime.h>
#include <stdint.h>

// AdaptiveEMA on MI455X (gfx1250, wave32).
// One 256-thread block per (b,f) series; chunked parallel scan over S=8192.
//   - t staged to LDS via async global->LDS copies (ASYNCcnt)
//   - decays computed once, cached in LDS (in-place over t)
//   - x loaded once into registers, reused for aggregate + emit passes
//   - y staged back into LDS (in-place over decays) and written out with
//     coalesced async LDS->global b128 stores

#define S_LEN   8192
#define NTHREADS 256
#define CHUNK   (S_LEN / NTHREADS)   // 32 elements per thread
#define QV      (CHUNK / 4)          // 8 float4 per thread
#define EMA_EPS 1e-10f

__device__ __forceinline__ float fast_exp2(float v) {
  return __builtin_amdgcn_exp2f(v);          // v_exp_f32 (base-2)
}
__device__ __forceinline__ float fast_log2(float v) {
  return __builtin_amdgcn_logf(v);           // v_log_f32 (base-2)
}

__device__ __forceinline__ void async_load_b128_to_lds(uint32_t lds_byte_off,
                                                       uint64_t gaddr) {
  // VDST = per-lane LDS byte address, VADDR = 64-bit global address (GV mode).
  asm volatile("global_load_async_to_lds_b128 %0, %1, off"
               :: "v"(lds_byte_off), "v"(gaddr)
               : "memory");
}

__device__ __forceinline__ void async_store_b128_from_lds(uint64_t gaddr,
                                                          uint32_t lds_byte_off) {
  // VADDR = 64-bit global address, VSRC = per-lane LDS byte address.
  asm volatile("global_store_async_from_lds_b128 %0, %1, off"
               :: "v"(gaddr), "v"(lds_byte_off)
               : "memory");
}

__device__ __forceinline__ void wait_asynccnt0() {
#if __has_builtin(__builtin_amdgcn_s_wait_asynccnt)
  __builtin_amdgcn_s_wait_asynccnt(0);
#else
  asm volatile("s_wait_asynccnt 0" ::: "memory");
#endif
}

__global__ void __launch_bounds__(NTHREADS)
AdaptiveEMA_34144990003442_kernel(const float* __restrict__ x,
                                  const float* __restrict__ t,
                                  const float* __restrict__ log_halflife,
                                  const float* __restrict__ power,
                                  const float* __restrict__ running_sum_dt,
                                  const float* __restrict__ dt_count,
                                  float* __restrict__ out,
                                  int F) {
  __shared__ __align__(16) float lds_td[S_LEN];   // t -> decay -> y (in place)
  __shared__ float sD[NTHREADS], sW[NTHREADS], sT[NTHREADS];

  const int tid    = threadIdx.x;
  const int series = blockIdx.x;           // series = b*F + f
  const int f      = series % F;
  const long base  = (long)series * S_LEN;

  // ---- per-feature constants (uniform across block) ----
  float pc = power[f];
  pc = fminf(fmaxf(pc, 1e-3f), 1.0f - 1e-3f);
  const float mean_dt  = running_sum_dt[f] / (dt_count[0] + EMA_EPS);
  const float inv_mean = 1.0f / mean_dt;
  // decay = exp(dt_pow * (-ln2/hl)) = exp2(dt_pow * (-1/hl)), hl = exp(log_hl)
  const float coef     = -__expf(-log_halflife[f]);
  const bool  is_sqrt  = (pc == 0.5f);     // power init is exactly 0.5

  const uint32_t lbase = (uint32_t)(uintptr_t)(&lds_td[0]);

  // ---- stage t[series, :] into LDS with async copies (ASYNCcnt) ----
  {
    const uint64_t gbase = (uint64_t)(uintptr_t)(t + base);
#pragma unroll
    for (int k = 0; k < QV; ++k) {                 // 8 iters, coalesced
      const uint32_t eb = (uint32_t)(k * NTHREADS + tid) * 16u;
      async_load_b128_to_lds(lbase + eb, gbase + eb);
    }
    wait_asynccnt0();
  }
  __syncthreads();

  const int start = tid * CHUNK;

  // boundary t value of the previous chunk (read before in-place overwrite)
  float prev_t = (tid > 0) ? lds_td[start - 1] : 0.0f;
  __syncthreads();

  // ---- phase 1: per-chunk decays + affine aggregate (D, W, T) ----
  float4 xq[QV];                                   // x held in registers
  {
    const float4* xg = (const float4*)(x + base + start);
#pragma unroll
    for (int q = 0; q < QV; ++q) xq[q] = xg[q];
  }

  float D = 1.0f, W = 0.0f, T = 0.0f;
#pragma unroll
  for (int q = 0; q < QV; ++q) {
    const float4 tv = *(const float4*)(&lds_td[start + q * 4]);
    const float ta[4] = {tv.x, tv.y, tv.z, tv.w};
    const float xa[4] = {xq[q].x, xq[q].y, xq[q].z, xq[q].w};
    float da[4];
#pragma unroll
    for (int u = 0; u < 4; ++u) {
      float dtv = ta[u] - prev_t;
      if (tid == 0 && q == 0 && u == 0) dtv = 1.0f;  // dt[0] = 1
      prev_t = ta[u];
      const float dtn  = dtv * inv_mean;
      const float dpow = is_sqrt ? __fsqrt_rn(dtn)
                                 : fast_exp2(pc * fast_log2(dtn));
      const float dec  = fast_exp2(dpow * coef);
      da[u] = dec;
      D *= dec;
      W = W * dec + xa[u];
      T = T * dec + 1.0f;
    }
    *(float4*)(&lds_td[start + q * 4]) = make_float4(da[0], da[1], da[2], da[3]);
  }

  // ---- phase 2: Hillis-Steele scan of chunk states across 256 threads ----
  sD[tid] = D; sW[tid] = W; sT[tid] = T;
  __syncthreads();
#pragma unroll
  for (int off = 1; off < NTHREADS; off <<= 1) {
    float d1 = 1.0f, w1 = 0.0f, t1 = 0.0f;
    const bool act = (tid >= off);
    if (act) { d1 = sD[tid - off]; w1 = sW[tid - off]; t1 = sT[tid - off]; }
    __syncthreads();
    if (act) {
      const float d2 = sD[tid], w2 = sW[tid], t2 = sT[tid];
      sD[tid] = d1 * d2;             // earlier ∘ later composition
      sW[tid] = w1 * d2 + w2;
      sT[tid] = t1 * d2 + t2;
    }
    __syncthreads();
  }
  float ws = 0.0f, tc = 0.0f;        // exclusive: state entering this chunk
  if (tid > 0) { ws = sW[tid - 1]; tc = sT[tid - 1]; }

  // ---- phase 3: replay chunk from incoming state; y overwrites decay ----
#pragma unroll
  for (int q = 0; q < QV; ++q) {
    const float4 dv = *(const float4*)(&lds_td[start + q * 4]);
    const float da[4] = {dv.x, dv.y, dv.z, dv.w};
    const float xa[4] = {xq[q].x, xq[q].y, xq[q].z, xq[q].w};
    float ya[4];
#pragma unroll
    for (int u = 0; u < 4; ++u) {
      ws = ws * da[u] + xa[u];
      tc = tc * da[u] + 1.0f;
      ya[u] = ws / (tc + EMA_EPS);
    }
    *(float4*)(&lds_td[start + q * 4]) = make_float4(ya[0], ya[1], ya[2], ya[3]);
  }
  __syncthreads();

  // ---- coalesced async LDS -> global stores of y ----
  {
    const uint64_t obase = (uint64_t)(uintptr_t)(out + base);
#pragma unroll
    for (int k = 0; k < QV; ++k) {                 // 8 iters, coalesced
      const uint32_t eb = (uint32_t)(k * NTHREADS + tid) * 16u;
      async_store_b128_from_lds(obase + eb, lbase + eb);
    }
    wait_asynccnt0();                // S_ENDPGM also implies wait-idle
  }
}

extern "C" void kernel_launch(void* const* d_in, const int* in_sizes, int n_in,
                              void* d_out, int out_size, void* d_ws, size_t ws_size,
                              hipStream_t stream) {
  (void)n_in; (void)d_ws; (void)ws_size; (void)out_size;
  const float* x              = (const float*)d_in[0];
  const float* t              = (const float*)d_in[1];
  const float* log_halflife   = (const float*)d_in[2];
  const float* power          = (const float*)d_in[3];
  const float* running_sum_dt = (const float*)d_in[4];
  const float* dt_count       = (const float*)d_in[5];
  float* out = (float*)d_out;

  const int F = in_sizes[2];                 // 256
  const int nseries = in_sizes[0] / S_LEN;   // B*F = 2048

  AdaptiveEMA_34144990003442_kernel<<<nseries, NTHREADS, 0, stream>>>(
      x, t, log_halflife, power, running_sum_dt, dt_count, out, F);
}